// MemoryBlock_24292335026522
// MI455X (gfx1250) — compile-verified
//
#include <hip/hip_runtime.h>
#include <math.h>

typedef float v2f __attribute__((ext_vector_type(2)));
typedef float v8f __attribute__((ext_vector_type(8)));

#define BB 256
#define LL 512
#define DD 256
#define HH 8
#define DH 32
#define GG 65
#define KK 8

__device__ __forceinline__ v8f wmma4(v2f a, v2f b, v8f c) {
    // V_WMMA_F32_16X16X4_F32: D = A(16x4) x B(4x16) + C
    return __builtin_amdgcn_wmma_f32_16x16x4_f32(
        /*neg_a=*/false, a, /*neg_b=*/false, b,
        /*c_mod=*/(short)0, c, /*reuse_a=*/false, /*reuse_b=*/false);
}

// -------------------------------------------------------------------------
// Y[M x 256] = X[M x Kdim] * W^T + bias, W is [256 x Kdim].
// Register-blocked: each wave computes 2 M-tiles x 4 N-tiles (2x4x16x16),
// so the k-loop does 6 fragment loads per 8 WMMA.
// Block = 8 waves arranged 4(M) x 2(N): block tile = 128 x 128.
// grid = (ceil(M/128), 2)
// -------------------------------------------------------------------------
__global__ __launch_bounds__(256) void gemm_xwT_bias(
    const float* __restrict__ X, int ldx,
    const float* __restrict__ W, int ldw,
    const float* __restrict__ bias,
    float* __restrict__ Y, int ldy,
    int M, int Kdim) {
    const int lane  = threadIdx.x & 31;
    const int wave  = threadIdx.x >> 5;
    const int mrow  = lane & 15;   // A row / B col
    const int khalf = lane >> 4;   // 0 -> K{0,1}, 1 -> K{2,3}
    const int mw = wave >> 1;      // 0..3
    const int nw = wave & 1;       // 0..1

    const int mBase = blockIdx.x * 128 + mw * 32;  // 2 M-tiles from here
    const int nBase = blockIdx.y * 128 + nw * 64;  // 4 N-tiles from here

    int r0 = mBase + mrow;       if (r0 >= M) r0 = M - 1;
    int r1 = mBase + 16 + mrow;  if (r1 >= M) r1 = M - 1;
    const float* xr0 = X + (size_t)r0 * ldx;
    const float* xr1 = X + (size_t)r1 * ldx;
    const float* wr0 = W + (size_t)(nBase +  0 + mrow) * ldw;
    const float* wr1 = W + (size_t)(nBase + 16 + mrow) * ldw;
    const float* wr2 = W + (size_t)(nBase + 32 + mrow) * ldw;
    const float* wr3 = W + (size_t)(nBase + 48 + mrow) * ldw;

    v8f acc[2][4] = {};
    for (int k0 = 0; k0 < Kdim; k0 += 4) {
        const int ko = k0 + 2 * khalf;
        v2f a0 = *(const v2f*)(xr0 + ko);
        v2f a1 = *(const v2f*)(xr1 + ko);
        v2f b0 = *(const v2f*)(wr0 + ko);
        v2f b1 = *(const v2f*)(wr1 + ko);
        v2f b2 = *(const v2f*)(wr2 + ko);
        v2f b3 = *(const v2f*)(wr3 + ko);
        acc[0][0] = wmma4(a0, b0, acc[0][0]);
        acc[0][1] = wmma4(a0, b1, acc[0][1]);
        acc[0][2] = wmma4(a0, b2, acc[0][2]);
        acc[0][3] = wmma4(a0, b3, acc[0][3]);
        acc[1][0] = wmma4(a1, b0, acc[1][0]);
        acc[1][1] = wmma4(a1, b1, acc[1][1]);
        acc[1][2] = wmma4(a1, b2, acc[1][2]);
        acc[1][3] = wmma4(a1, b3, acc[1][3]);
    }
#pragma unroll
    for (int j = 0; j < 4; ++j) {
        const int col = nBase + j * 16 + mrow;
        const float bv = bias ? bias[col] : 0.0f;
#pragma unroll
        for (int mi = 0; mi < 2; ++mi) {
            const int baseRow = mBase + mi * 16 + 8 * khalf;
#pragma unroll
            for (int i = 0; i < 8; ++i) {
                const int r = baseRow + i;
                if (r < M) Y[(size_t)r * ldy + col] = acc[mi][j][i] + bv;
            }
        }
    }
}

// -------------------------------------------------------------------------
// Fused K/V projection: kbuf = x*Wk^T + bk, vbuf = x*Wv^T + bv.
// Each wave: 2 M-tiles x 2 N-tiles x {K,V} (8 accumulators), sharing the
// x A-fragments across both weight matrices: 6 loads per 8 WMMA.
// Block = 8 waves arranged 4(M) x 2(N): block tile = 128 rows x 64 cols.
// grid = (B*L/128, 4). M = B*L = 131072, no store guards needed.
// -------------------------------------------------------------------------
__global__ __launch_bounds__(256) void kv_proj(
    const float* __restrict__ x,
    const float* __restrict__ Wk, const float* __restrict__ bk,
    const float* __restrict__ Wv, const float* __restrict__ bv,
    float* __restrict__ kbuf, float* __restrict__ vbuf) {
    const int lane  = threadIdx.x & 31;
    const int wave  = threadIdx.x >> 5;
    const int mrow  = lane & 15;
    const int khalf = lane >> 4;
    const int mw = wave & 3;       // 0..3
    const int nw = wave >> 2;      // 0..1

    const int mBase = blockIdx.x * 128 + mw * 32;
    const int nBase = blockIdx.y * 64 + nw * 32;

    const float* xr0 = x + (size_t)(mBase + mrow) * DD;
    const float* xr1 = x + (size_t)(mBase + 16 + mrow) * DD;
    const float* wk0 = Wk + (size_t)(nBase +  0 + mrow) * DD;
    const float* wk1 = Wk + (size_t)(nBase + 16 + mrow) * DD;
    const float* wv0 = Wv + (size_t)(nBase +  0 + mrow) * DD;
    const float* wv1 = Wv + (size_t)(nBase + 16 + mrow) * DD;

    v8f ka[2][2] = {};
    v8f va[2][2] = {};
    for (int k0 = 0; k0 < DD; k0 += 4) {
        const int ko = k0 + 2 * khalf;
        v2f a0 = *(const v2f*)(xr0 + ko);
        v2f a1 = *(const v2f*)(xr1 + ko);
        v2f k0f = *(const v2f*)(wk0 + ko);
        v2f k1f = *(const v2f*)(wk1 + ko);
        v2f v0f = *(const v2f*)(wv0 + ko);
        v2f v1f = *(const v2f*)(wv1 + ko);
        ka[0][0] = wmma4(a0, k0f, ka[0][0]);
        ka[0][1] = wmma4(a0, k1f, ka[0][1]);
        ka[1][0] = wmma4(a1, k0f, ka[1][0]);
        ka[1][1] = wmma4(a1, k1f, ka[1][1]);
        va[0][0] = wmma4(a0, v0f, va[0][0]);
        va[0][1] = wmma4(a0, v1f, va[0][1]);
        va[1][0] = wmma4(a1, v0f, va[1][0]);
        va[1][1] = wmma4(a1, v1f, va[1][1]);
    }
#pragma unroll
    for (int j = 0; j < 2; ++j) {
        const int col = nBase + j * 16 + mrow;
        const float bkv = bk[col];
        const float bvv = bv[col];
#pragma unroll
        for (int mi = 0; mi < 2; ++mi) {
            const int baseRow = mBase + mi * 16 + 8 * khalf;
#pragma unroll
            for (int i = 0; i < 8; ++i) {
                const size_t off = (size_t)(baseRow + i) * DD + col;
                kbuf[off] = ka[mi][j][i] + bkv;
                vbuf[off] = va[mi][j][i] + bvv;
            }
        }
    }
}

// -------------------------------------------------------------------------
// Fused scores + top-8 + gather + grouped conv.
// One block per (b, h, 16-query tile). Softmax/scale skipped: both are
// monotonic, so top_k indices of the raw logits match the reference's
// top_k of softmax. Ties break to the lowest index (strict '>' scan +
// lowest-index-wins reduction), matching jax.lax.top_k.
// -------------------------------------------------------------------------
#define SC_LD 516   // LDS row stride: 8*516 % 64 == 32 -> no bank conflicts
__global__ __launch_bounds__(256) void attn_topk(
    const float* __restrict__ qbuf,   // [G, 256]
    const float* __restrict__ kbuf,   // [B*L, 256]
    const float* __restrict__ vbuf,   // [B*L, 256]
    const float* __restrict__ gw,     // [G, K]
    const float* __restrict__ gb,     // [G]
    float* __restrict__ aggbuf) {     // [B*G, 256]
    __shared__ float sc[16][SC_LD];

    const int gt = blockIdx.x;        // 0..4
    const int h  = blockIdx.y;        // 0..7
    const int b  = blockIdx.z;        // 0..255
    const int lane  = threadIdx.x & 31;
    const int wave  = threadIdx.x >> 5;
    const int mrow  = lane & 15;
    const int khalf = lane >> 4;
    const int g0 = gt * 16;

    // ---- score tile [16 x 512] via WMMA: Q(16x32) @ K^T(32x512) ----
    int qrow = g0 + mrow;
    if (qrow >= GG) qrow = GG - 1;
    const float* qr = qbuf + (size_t)qrow * DD + h * DH;
    v2f afr[8];
#pragma unroll
    for (int k = 0; k < 8; ++k) afr[k] = *(const v2f*)(qr + 4 * k + 2 * khalf);

    const float* kb = kbuf + (size_t)b * LL * DD + h * DH;
    for (int t = 0; t < 4; ++t) {
        const int ntile = wave * 4 + t;                 // 0..31
        const float* kr = kb + (size_t)(ntile * 16 + mrow) * DD;
        v8f acc = {};
#pragma unroll
        for (int k = 0; k < 8; ++k) {
            v2f bfr = *(const v2f*)(kr + 4 * k + 2 * khalf);
            acc = wmma4(afr[k], bfr, acc);
        }
#pragma unroll
        for (int i = 0; i < 8; ++i)
            sc[i + 8 * khalf][ntile * 16 + mrow] = acc[i];
    }
    __syncthreads();

    // ---- per-row top-8 (rows are wave-exclusive: wave handles r, r+8) ----
    for (int rsel = 0; rsel < 2; ++rsel) {
        const int r = wave + 8 * rsel;
        const int g = g0 + r;
        float* row = sc[r];
        int idxs[KK];
#pragma unroll 1
        for (int p = 0; p < KK; ++p) {
            // lane scans its contiguous 16 elements; strict '>' keeps lowest idx
            float bvv = -INFINITY;
            int   bii = lane * 16;
#pragma unroll
            for (int j = 0; j < 16; ++j) {
                float v = row[lane * 16 + j];
                if (v > bvv) { bvv = v; bii = lane * 16 + j; }
            }
            // wave32 argmax reduction, tie -> lowest index
            for (int off = 16; off > 0; off >>= 1) {
                float ov = __shfl_xor(bvv, off, 32);
                int   oi = __shfl_xor(bii, off, 32);
                if (ov > bvv || (ov == bvv && oi < bii)) { bvv = ov; bii = oi; }
            }
            idxs[p] = bii;
            if ((bii >> 4) == lane) row[bii] = -INFINITY;  // remove winner
        }
        // ---- gather V at top-8 and apply grouped conv (kernel=1) ----
        if (g < GG) {
            float a = gb[g];
#pragma unroll
            for (int p = 0; p < KK; ++p) {
                const float* vr = vbuf + (size_t)(b * LL + idxs[p]) * DD + h * DH;
                a += gw[g * KK + p] * vr[lane];
            }
            aggbuf[(size_t)(b * GG + g) * DD + h * DH + lane] = a;
        }
    }
}

// -------------------------------------------------------------------------
extern "C" void kernel_launch(void* const* d_in, const int* in_sizes, int n_in,
                              void* d_out, int out_size, void* d_ws, size_t ws_size,
                              hipStream_t stream) {
    const float* x   = (const float*)d_in[0];
    const float* tgt = (const float*)d_in[1];
    const float* Wq  = (const float*)d_in[2];
    const float* bq  = (const float*)d_in[3];
    const float* Wk  = (const float*)d_in[4];
    const float* bk  = (const float*)d_in[5];
    const float* Wv  = (const float*)d_in[6];
    const float* bv  = (const float*)d_in[7];
    const float* gw  = (const float*)d_in[8];
    const float* gb  = (const float*)d_in[9];
    const float* Wo  = (const float*)d_in[10];
    const float* bo  = (const float*)d_in[11];
    float* out = (float*)d_out;

    // workspace layout
    char* ws = (char*)d_ws;
    const size_t kvBytes = (size_t)BB * LL * DD * sizeof(float);   // 134 MB each
    float* qbuf   = (float*)ws;                                    // 65*256*4
    float* kbuf   = (float*)(ws + (1u << 20));
    float* vbuf   = (float*)(ws + (1u << 20) + kvBytes);
    float* aggbuf = (float*)(ws + (1u << 20) + 2 * kvBytes);       // 16640*256*4

    // 1) Q projection: [65,256] = tgt * Wq^T + bq   (1 block row, clamped)
    gemm_xwT_bias<<<dim3(1, 2), 256, 0, stream>>>(
        tgt, DD, Wq, DD, bq, qbuf, DD, GG, DD);

    // 2) fused K/V projections: [B*L, 256]
    kv_proj<<<dim3((BB * LL) / 128, 4), 256, 0, stream>>>(
        x, Wk, bk, Wv, bv, kbuf, vbuf);

    // 3) scores + top-8 + gather + grouped conv -> aggbuf [B*G, 256]
    attn_topk<<<dim3(5, HH, BB), 256, 0, stream>>>(
        qbuf, kbuf, vbuf, gw, gb, aggbuf);

    // 4) output projection: out[B*G, 256] = aggbuf * Wo^T + bo
    gemm_xwT_bias<<<dim3((BB * GG) / 128, 2), 256, 0, stream>>>(
        aggbuf, DD, Wo, DD, bo, out, DD, BB * GG, DD);
}